// SpinorBilinears_79370995631003
// MI455X (gfx1250) — compile-verified
//
#include <hip/hip_runtime.h>
#include <hip/hip_bf16.h>

typedef __attribute__((ext_vector_type(2))) float v2f;
typedef __attribute__((ext_vector_type(8))) float v8f;

// D = A(16x4) * B(4x16) + C, f32 WMMA (exact f32 arithmetic, matches reference)
static __device__ inline v8f wmma_f32x4(v2f a, v2f b, v8f c) {
    return __builtin_amdgcn_wmma_f32_16x16x4_f32(
        /*neg_a=*/false, a, /*neg_b=*/false, b,
        /*c_mod=*/(short)0, c, /*reuse_a=*/false, /*reuse_b=*/false);
}

// Token-tiled bilinear contraction:
//   K0[T]      = psi.psi
//   K1[T,c,d]  = psi^T * 0.5(Gcd - Gdc) * psi   (Gcd = gamma_c gamma_d)
//   K2[T,c,d]  = psi^T * 0.5(Gcd + Gdc) * psi
// GEMM view per 16-token tile: [M=16 tokens x K=16(ij)] x [K=16 x N=64(cd)]
__global__ __launch_bounds__(256) void SpinorBilinears_wmma_kernel(
    const float* __restrict__ psi,     // [tokens, 4]
    const float* __restrict__ gamma,   // [8, 4, 4]
    float* __restrict__ outK0,         // [tokens]
    float* __restrict__ outK1,         // [tokens, 64]
    float* __restrict__ outK2,         // [tokens, 64]
    int numTiles)
{
    const int lane = threadIdx.x & 31;
    const int n    = lane & 15;   // N index within 16-wide tile; token row within half
    const int h    = lane >> 4;   // half: K cols {2h,2h+1}; D rows offset 8h

    // ---------------- constant B-fragments from gamma (once per wave) -------------
    // bX[nt][kc][js] = Mat[cd = nt*16+n][ i=kc, j=2h+js ]
    float bA[4][4][2];
    float bS[4][4][2];
    #pragma unroll
    for (int nt = 0; nt < 4; ++nt) {
        const int cd = nt * 16 + n;
        const int c  = cd >> 3;
        const int d  = cd & 7;
        float gc[16], gd[16];
        #pragma unroll
        for (int q = 0; q < 16; ++q) { gc[q] = gamma[c * 16 + q]; gd[q] = gamma[d * 16 + q]; }
        #pragma unroll
        for (int i = 0; i < 4; ++i) {
            #pragma unroll
            for (int js = 0; js < 2; ++js) {
                const int j = 2 * h + js;
                float g1 = 0.f, g2 = 0.f;        // Gcd[i,j], Gdc[i,j]
                #pragma unroll
                for (int t = 0; t < 4; ++t) {
                    g1 += gc[i * 4 + t] * gd[t * 4 + j];
                    g2 += gd[i * 4 + t] * gc[t * 4 + j];
                }
                bA[nt][i][js] = 0.5f * (g1 - g2);
                bS[nt][i][js] = 0.5f * (g1 + g2);
            }
        }
    }

    const int gtid   = blockIdx.x * blockDim.x + threadIdx.x;
    const int wave   = gtid >> 5;
    const int nWaves = (gridDim.x * blockDim.x) >> 5;

    for (int tile = wave; tile < numTiles; tile += nWaves) {
        const int T0  = tile << 4;
        const int tok = T0 + n;                       // both halves: same token's psi
        const float4 p = ((const float4*)psi)[tok];   // global_load_b128

        // K0: all 32 lanes store (halves write identical values -> EXEC stays full)
        __builtin_nontemporal_store(p.x * p.x + p.y * p.y + p.z * p.z + p.w * p.w,
                                    outK0 + tok);

        // A-fragments: outer[i=kc, j=2h+slot] = psi[kc] * psi[2h+slot]
        const float pk[4] = {p.x, p.y, p.z, p.w};
        const float pj0 = h ? p.z : p.x;
        const float pj1 = h ? p.w : p.y;
        v2f aFrag[4];
        #pragma unroll
        for (int kc = 0; kc < 4; ++kc) {
            aFrag[kc].x = pk[kc] * pj0;
            aFrag[kc].y = pk[kc] * pj1;
        }

        // Single base address per tile; every store below uses a constant
        // byte offset (max (7*64+3*16)*4 = 1984 B) -> immediate IOFFSET forms.
        const unsigned baseElem = (unsigned)(T0 + 8 * h) * 64u + (unsigned)n;
        float* __restrict__ p1 = outK1 + baseElem;
        float* __restrict__ p2 = outK2 + baseElem;

        #pragma unroll
        for (int nt = 0; nt < 4; ++nt) {
            v8f accA = {0.f, 0.f, 0.f, 0.f, 0.f, 0.f, 0.f, 0.f};
            v8f accS = {0.f, 0.f, 0.f, 0.f, 0.f, 0.f, 0.f, 0.f};
            #pragma unroll
            for (int kc = 0; kc < 4; ++kc) {
                v2f ba; ba.x = bA[nt][kc][0]; ba.y = bA[nt][kc][1];
                v2f bs; bs.x = bS[nt][kc][0]; bs.y = bS[nt][kc][1];
                accA = wmma_f32x4(aFrag[kc], ba, accA);
                accS = wmma_f32x4(aFrag[kc], bs, accS);
            }
            // D layout: VGPR r, lane -> (M = r + 8h, N = n); row stride 64 elems
            #pragma unroll
            for (int r = 0; r < 8; ++r) {
                __builtin_nontemporal_store(accA[r], p1 + r * 64 + nt * 16);
                __builtin_nontemporal_store(accS[r], p2 + r * 64 + nt * 16);
            }
        }
    }
}

extern "C" void kernel_launch(void* const* d_in, const int* in_sizes, int n_in,
                              void* d_out, int out_size, void* d_ws, size_t ws_size,
                              hipStream_t stream) {
    const float* psi   = (const float*)d_in[0];   // [B, N, 4] f32
    const float* gamma = (const float*)d_in[1];   // [8, 4, 4] f32

    const long tokens = (long)in_sizes[0] / 4;    // B*N = 524288
    float* outK0 = (float*)d_out;
    float* outK1 = outK0 + tokens;
    float* outK2 = outK1 + tokens * 64;

    const int numTiles = (int)(tokens >> 4);      // 16 tokens per WMMA tile
    const int threads  = 256;                     // 8 wave32 per block
    int blocks = 1024;                            // ~8192 waves, 4 tiles each
    const int wavesNeeded = numTiles;             // at most one tile per wave needed
    if (blocks * (threads / 32) > wavesNeeded) {
        blocks = (wavesNeeded * 32 + threads - 1) / threads;
        if (blocks < 1) blocks = 1;
    }

    SpinorBilinears_wmma_kernel<<<blocks, threads, 0, stream>>>(
        psi, gamma, outK0, outK1, outK2, numTiles);
}